// GraphTransformerLayer_60017872994810
// MI455X (gfx1250) — compile-verified
//
#include <hip/hip_runtime.h>
#include <math.h>
#include <stdint.h>

// ---------------------------------------------------------------------------
// GraphTransformerLayer for MI455X (gfx1250), fp32 WMMA path.
//
// Math reduction: head_dim == 1  =>  softmax over singleton axis == 1.0
//   =>  attn == V == x@Wv + bv   (Q, K, edge_index are dead).
//   h   = LN(x + x@Wv + bv)*g1 + be1
//   out = LN(h + gelu(h@W1+bf1)@W2 + bf2)*g2 + be2
//
// R=32 row blocks: weight reads are compulsory per block, so traffic is
// (16384/R) * 36MB of L2->WGP bandwidth; R=32 halves it vs R=16 and gives
// each B fragment 2 WMMAs of reuse (register blocking in M).
// ---------------------------------------------------------------------------

typedef __attribute__((ext_vector_type(2))) float v2f;
typedef __attribute__((ext_vector_type(8))) float v8f;
typedef int v4i __attribute__((vector_size(16)));   // matches builtin prototype
typedef __attribute__((address_space(3))) v4i lds_v4i;

#define LDA 1028   // padded LDS row stride (floats) for 1024-wide tiles
#define LDT 132    // padded LDS row stride (floats) for 128-wide t tiles
#define ROWS 16384 // B*N
#define DIM 1024
#define DFF 4096
#define RB 32      // rows per block

#if __has_builtin(__builtin_amdgcn_global_load_async_to_lds_b128) && \
    __has_builtin(__builtin_amdgcn_s_wait_asynccnt)
#define USE_ASYNC_LDS 1
#else
#define USE_ASYNC_LDS 0
#endif

__device__ __forceinline__ v8f wmma_f32(v2f a, v2f b, v8f c) {
  // V_WMMA_F32_16X16X4_F32: D(16x16,f32) = A(16x4,f32) * B(4x16,f32) + C
  return __builtin_amdgcn_wmma_f32_16x16x4_f32(
      false, a, false, b, (short)0, c, false, false);
}

__device__ __forceinline__ v8f zero8() {
  v8f z;
#pragma unroll
  for (int i = 0; i < 8; ++i) z[i] = 0.0f;
  return z;
}

__device__ __forceinline__ float gelu_exact(float x) {
  // gelu(approximate=False): 0.5*x*(1+erf(x/sqrt(2)))
  return 0.5f * x * (1.0f + erff(x * 0.70710678118654752440f));
}

// Preload a 32 x 1024 fp32 tile (rows row0..row0+31) into LDS (stride LDA).
// Uses GLOBAL_LOAD_ASYNC_TO_LDS_B128 (ASYNCcnt) when available.
__device__ __forceinline__ void load_tile32(const float* __restrict__ G,
                                            int row0, float* sA, int tid) {
  const float4* gx = (const float4*)(G + (size_t)row0 * DIM);
#if USE_ASYNC_LDS
#pragma unroll 8
  for (int it = 0; it < 32; ++it) {
    int j  = tid + it * 256;   // float4 index, 0..8191
    int r  = j >> 8;           // row 0..31
    int c4 = j & 255;          // float4 column
    v4i* gp = (v4i*)(uintptr_t)(gx + r * 256 + c4);
    lds_v4i* lp =
        (lds_v4i*)(uint32_t)(uintptr_t)(sA + r * LDA + c4 * 4);
    __builtin_amdgcn_global_load_async_to_lds_b128(gp, lp, 0, 0);
  }
  __builtin_amdgcn_s_wait_asynccnt(0);
#else
#pragma unroll 8
  for (int it = 0; it < 32; ++it) {
    int j  = tid + it * 256;
    int r  = j >> 8;
    int c4 = j & 255;
    *(float4*)(sA + r * LDA + c4 * 4) = gx[r * 256 + c4];
  }
#endif
}

// ---------------------------------------------------------------------------
// Kernel 1: h = LayerNorm(x + x@Wv + bv) * g1 + be1
// grid = 512 blocks (32 rows each), block = 256 threads (8 waves).
// Wave w owns cols [w*128, w*128+128) x all 32 rows: 2 M-tiles x 8 N-tiles.
// ---------------------------------------------------------------------------
__global__ __launch_bounds__(256) void gt_vproj_ln1(
    const float* __restrict__ X, const float* __restrict__ Wv,
    const float* __restrict__ bv, const float* __restrict__ g1,
    const float* __restrict__ be1, float* __restrict__ H) {
  extern __shared__ float smem[];
  float* sA   = smem;               // RB*LDA floats
  float* sSum = smem + RB * LDA;    // RB floats
  float* sSq  = sSum + RB;          // RB floats

  const int tid  = threadIdx.x;
  const int wave = tid >> 5;
  const int lane = tid & 31;
  const int lm   = lane & 15;
  const int lh   = lane >> 4;
  const int row0 = blockIdx.x * RB;

  if (tid < RB) sSum[tid] = 0.0f;
  else if (tid < 2 * RB) sSq[tid - RB] = 0.0f;

  load_tile32(X, row0, sA, tid);
  __syncthreads();

  const int n0 = wave * 128;
  v8f acc[2][8];
#pragma unroll
  for (int mt = 0; mt < 2; ++mt)
#pragma unroll
    for (int t = 0; t < 8; ++t) acc[mt][t] = zero8();

  const float* Arow0 = sA + lm * LDA + 2 * lh;        // rows 0..15
  const float* Arow1 = Arow0 + 16 * LDA;              // rows 16..31
  const float* Bbase = Wv + (size_t)(2 * lh) * DIM + n0 + lm;

#pragma unroll 2
  for (int k0 = 0; k0 < DIM; k0 += 4) {
    v2f a0 = *(const v2f*)(Arow0 + k0);
    v2f a1 = *(const v2f*)(Arow1 + k0);
    const float* bp = Bbase + (size_t)k0 * DIM;
#pragma unroll
    for (int t = 0; t < 8; ++t) {
      v2f b;
      b.x = bp[t * 16];
      b.y = bp[t * 16 + DIM];
      acc[0][t] = wmma_f32(a0, b, acc[0][t]);
      acc[1][t] = wmma_f32(a1, b, acc[1][t]);
    }
  }

  // Epilogue: + bias + residual(x), then per-row statistics for LayerNorm.
  // C layout: tile (mt,t), vgpr r, lane -> row m = mt*16 + r + 8*lh,
  //                                       col n = n0 + 16*t + lm.
#pragma unroll
  for (int t = 0; t < 8; ++t) {
    int n = n0 + t * 16 + lm;
    float bb = bv[n];
#pragma unroll
    for (int mt = 0; mt < 2; ++mt)
#pragma unroll
      for (int r = 0; r < 8; ++r) {
        int m = mt * 16 + r + 8 * lh;
        acc[mt][t][r] += bb + sA[m * LDA + n];
      }
  }
#pragma unroll
  for (int mt = 0; mt < 2; ++mt)
#pragma unroll
    for (int r = 0; r < 8; ++r) {
      int m = mt * 16 + r + 8 * lh;
      float s = 0.0f, q = 0.0f;
#pragma unroll
      for (int t = 0; t < 8; ++t) {
        float v = acc[mt][t][r];
        s += v;
        q += v * v;
      }
      atomicAdd(&sSum[m], s);
      atomicAdd(&sSq[m], q);
    }
  __syncthreads();

  const float inv = 1.0f / (float)DIM;
#pragma unroll
  for (int t = 0; t < 8; ++t) {
    int n = n0 + t * 16 + lm;
    float gg = g1[n], ee = be1[n];
#pragma unroll
    for (int mt = 0; mt < 2; ++mt)
#pragma unroll
      for (int r = 0; r < 8; ++r) {
        int m = mt * 16 + r + 8 * lh;
        float mu   = sSum[m] * inv;
        float var  = sSq[m] * inv - mu * mu;
        float rstd = rsqrtf(var + 1e-5f);
        H[(size_t)(row0 + m) * DIM + n] = (acc[mt][t][r] - mu) * rstd * gg + ee;
      }
  }
}

// ---------------------------------------------------------------------------
// Kernel 2: out = LayerNorm(h + gelu(h@W1+bf1)@W2 + bf2) * g2 + be2
// Dff processed in 32 chunks of 128; GELU intermediate staged through LDS.
// ---------------------------------------------------------------------------
__global__ __launch_bounds__(256) void gt_ffn_ln2(
    const float* __restrict__ H, const float* __restrict__ W1,
    const float* __restrict__ bf1, const float* __restrict__ W2,
    const float* __restrict__ bf2, const float* __restrict__ g2,
    const float* __restrict__ be2, float* __restrict__ Out) {
  extern __shared__ float smem[];
  float* sA   = smem;                    // RB*LDA floats: h tile
  float* sT   = smem + RB * LDA;         // RB*LDT floats: gelu chunk
  float* sSum = sT + RB * LDT;           // RB floats
  float* sSq  = sSum + RB;               // RB floats

  const int tid  = threadIdx.x;
  const int wave = tid >> 5;
  const int lane = tid & 31;
  const int lm   = lane & 15;
  const int lh   = lane >> 4;
  const int row0 = blockIdx.x * RB;

  if (tid < RB) sSum[tid] = 0.0f;
  else if (tid < 2 * RB) sSq[tid - RB] = 0.0f;

  load_tile32(H, row0, sA, tid);
  __syncthreads();

  const int n0 = wave * 128;
  v8f acc[2][8];
#pragma unroll
  for (int mt = 0; mt < 2; ++mt)
#pragma unroll
    for (int t = 0; t < 8; ++t) acc[mt][t] = zero8();

  const float* Arow0  = sA + lm * LDA + 2 * lh;
  const float* Arow1  = Arow0 + 16 * LDA;
  const float* Trow0  = sT + lm * LDT + 2 * lh;
  const float* Trow1  = Trow0 + 16 * LDT;
  const float* B1base = W1 + (size_t)(2 * lh) * DFF + lm;       // + chunk col
  const float* B2base = W2 + (size_t)(2 * lh) * DIM + n0 + lm;  // + chunk*DIM

  for (int ch = 0; ch < 32; ++ch) {
    const int c0 = ch * 128;

    // ---- Stage 1: t[32 x 16w..16w+16] = h @ W1[:, chunk], K=1024 ----
    v8f tacc[2];
    tacc[0] = zero8();
    tacc[1] = zero8();
    const float* bp1 = B1base + c0 + wave * 16;
#pragma unroll 4
    for (int k0 = 0; k0 < DIM; k0 += 4) {
      v2f a0 = *(const v2f*)(Arow0 + k0);
      v2f a1 = *(const v2f*)(Arow1 + k0);
      const float* p = bp1 + (size_t)k0 * DFF;
      v2f b;
      b.x = p[0];
      b.y = p[DFF];
      tacc[0] = wmma_f32(a0, b, tacc[0]);
      tacc[1] = wmma_f32(a1, b, tacc[1]);
    }
    float bb1 = bf1[c0 + wave * 16 + lm];

    __syncthreads();  // all waves done reading sT from previous chunk
#pragma unroll
    for (int mt = 0; mt < 2; ++mt)
#pragma unroll
      for (int r = 0; r < 8; ++r) {
        int m = mt * 16 + r + 8 * lh;
        sT[m * LDT + wave * 16 + lm] = gelu_exact(tacc[mt][r] + bb1);
      }
    __syncthreads();  // sT chunk ready

    // ---- Stage 2: acc += t(32x128) @ W2[c0:c0+128, n0:n0+128] ----
    const float* bp2 = B2base + (size_t)c0 * DIM;
#pragma unroll 2
    for (int kl = 0; kl < 128; kl += 4) {
      v2f a0 = *(const v2f*)(Trow0 + kl);
      v2f a1 = *(const v2f*)(Trow1 + kl);
      const float* p = bp2 + (size_t)kl * DIM;
#pragma unroll
      for (int t = 0; t < 8; ++t) {
        v2f b;
        b.x = p[t * 16];
        b.y = p[t * 16 + DIM];
        acc[0][t] = wmma_f32(a0, b, acc[0][t]);
        acc[1][t] = wmma_f32(a1, b, acc[1][t]);
      }
    }
  }

  // Epilogue: + bf2 + residual(h), LayerNorm2, write out.
#pragma unroll
  for (int t = 0; t < 8; ++t) {
    int n = n0 + t * 16 + lm;
    float bb = bf2[n];
#pragma unroll
    for (int mt = 0; mt < 2; ++mt)
#pragma unroll
      for (int r = 0; r < 8; ++r) {
        int m = mt * 16 + r + 8 * lh;
        acc[mt][t][r] += bb + sA[m * LDA + n];
      }
  }
#pragma unroll
  for (int mt = 0; mt < 2; ++mt)
#pragma unroll
    for (int r = 0; r < 8; ++r) {
      int m = mt * 16 + r + 8 * lh;
      float s = 0.0f, q = 0.0f;
#pragma unroll
      for (int t = 0; t < 8; ++t) {
        float v = acc[mt][t][r];
        s += v;
        q += v * v;
      }
      atomicAdd(&sSum[m], s);
      atomicAdd(&sSq[m], q);
    }
  __syncthreads();

  const float inv = 1.0f / (float)DIM;
#pragma unroll
  for (int t = 0; t < 8; ++t) {
    int n = n0 + t * 16 + lm;
    float gg = g2[n], ee = be2[n];
#pragma unroll
    for (int mt = 0; mt < 2; ++mt)
#pragma unroll
      for (int r = 0; r < 8; ++r) {
        int m = mt * 16 + r + 8 * lh;
        float mu   = sSum[m] * inv;
        float var  = sSq[m] * inv - mu * mu;
        float rstd = rsqrtf(var + 1e-5f);
        Out[(size_t)(row0 + m) * DIM + n] = (acc[mt][t][r] - mu) * rstd * gg + ee;
      }
  }
}

// ---------------------------------------------------------------------------
extern "C" void kernel_launch(void* const* d_in, const int* in_sizes, int n_in,
                              void* d_out, int out_size, void* d_ws, size_t ws_size,
                              hipStream_t stream) {
  const float* x   = (const float*)d_in[0];
  // d_in[1] edge_index: unused by the reference computation.
  // d_in[2..5] Wq,bq,Wk,bk: dead (softmax over head_dim==1 is identically 1).
  const float* Wv  = (const float*)d_in[6];
  const float* bv  = (const float*)d_in[7];
  const float* g1  = (const float*)d_in[8];
  const float* be1 = (const float*)d_in[9];
  const float* W1  = (const float*)d_in[10];
  const float* bf1 = (const float*)d_in[11];
  const float* W2  = (const float*)d_in[12];
  const float* bf2 = (const float*)d_in[13];
  const float* g2  = (const float*)d_in[14];
  const float* be2 = (const float*)d_in[15];
  float* out = (float*)d_out;

  // h buffer: 64MB. Prefer scratch; aliasing d_out is safe because each
  // kernel-2 block reads only its own 32 rows of h (into LDS) before writing
  // those same 32 rows of out.
  const size_t hbytes = (size_t)ROWS * DIM * sizeof(float);
  float* h = (ws_size >= hbytes) ? (float*)d_ws : out;

  const dim3 grid(ROWS / RB), block(256);
  const size_t lds1 = (RB * LDA + 2 * RB) * sizeof(float);            // ~131.8 KB
  const size_t lds2 = (RB * LDA + RB * LDT + 2 * RB) * sizeof(float); // ~148.7 KB

  (void)hipFuncSetAttribute((const void*)gt_vproj_ln1,
                            hipFuncAttributeMaxDynamicSharedMemorySize, (int)lds1);
  (void)hipFuncSetAttribute((const void*)gt_ffn_ln2,
                            hipFuncAttributeMaxDynamicSharedMemorySize, (int)lds2);

  gt_vproj_ln1<<<grid, block, lds1, stream>>>(x, Wv, bv, g1, be1, h);
  gt_ffn_ln2<<<grid, block, lds2, stream>>>(h, W1, bf1, W2, bf2, g2, be2, out);
}